// DigitCaps_481036337879
// MI455X (gfx1250) — compile-verified
//
#include <hip/hip_runtime.h>
#include <math.h>

// Problem constants (match reference)
#define C_CAPS   10
#define B_SZ     512
#define N_NODES  1152
#define I_CH     8
#define O_CH     16
#define K_TOT    (N_NODES * I_CH)   // 9216
#define BT       16                  // batch tile (WMMA M)
#define NWAVES   8
#define NTHREADS (NWAVES * 32)
#define N_PER_WAVE (N_NODES / NWAVES) // 144
#define SN       1153                 // padded LDS row stride (stride 1 mod 64 banks)

typedef float v2f __attribute__((ext_vector_type(2)));
typedef float v8f __attribute__((ext_vector_type(8)));

// Reduce over the 16-lane half of a wave32 (masks 1,2,4,8 stay inside each half)
__device__ __forceinline__ float half16_sum(float v) {
    v += __shfl_xor(v, 1, 32);
    v += __shfl_xor(v, 2, 32);
    v += __shfl_xor(v, 4, 32);
    v += __shfl_xor(v, 8, 32);
    return v;
}
__device__ __forceinline__ float half16_max(float v) {
    v = fmaxf(v, __shfl_xor(v, 1, 32));
    v = fmaxf(v, __shfl_xor(v, 2, 32));
    v = fmaxf(v, __shfl_xor(v, 4, 32));
    v = fmaxf(v, __shfl_xor(v, 8, 32));
    return v;
}

// One workgroup = one (capsule c, 16-batch tile). u_hat is never materialized:
// it is recomputed from L2-resident u/W via chained v_wmma_f32_16x16x4_f32.
__global__ __launch_bounds__(NTHREADS)
void digitcaps_routing_kernel(const float* __restrict__ u,   // [B, N, I]
                              const float* __restrict__ W,   // [C, N, I, O]
                              float* __restrict__ out)       // [C, B, 1, 1, O]
{
    __shared__ float a_lds[BT * SN];          // routing logits b[b,n]  (~73.8 KB)
    __shared__ float red[NWAVES][BT * O_CH];  // per-wave GEMM partials  (8 KB)
    __shared__ float v_lds[BT * O_CH];        // v_j per (b,o)           (1 KB)
    __shared__ float amax_s[BT];
    __shared__ float invz_s[BT];

    const int c    = blockIdx.y;
    const int bb   = blockIdx.x * BT;        // batch base of this tile
    const int t    = threadIdx.x;
    const int wv   = t >> 5;                 // wave id 0..7
    const int ln   = t & 31;                 // lane in wave32
    const int half = ln >> 4;                // 0: lanes 0-15, 1: lanes 16-31
    const int l16  = ln & 15;                // A row m / B,C,D column o

    const float* __restrict__ Wc = W + (size_t)c * (N_NODES * I_CH * O_CH);
    const float* __restrict__ Ub = u + (size_t)bb * K_TOT;

    const int n0 = wv * N_PER_WAVE;          // this wave's n-range [n0, n0+144)

    // ---- accumulated GEMM pass: red[wv] = sum_n scale(b,n) * u_hat_tile(n) ----
    // weighted==false: scale = 1/N (iteration 0 uniform coupling)
    // weighted==true : scale = softmax logits via amax_s/invz_s + a_lds
    auto gemm_pass = [&](bool weighted) {
        v8f acc = {};
        const float am = weighted ? amax_s[l16] : 0.0f;
        const float iz = weighted ? invz_s[l16] : 0.0f;
        for (int n = n0; n < n0 + N_PER_WAVE; ++n) {
            float scale;
            if (weighted)
                scale = __expf(a_lds[l16 * SN + n] - am) * iz;  // c_ij[b=l16, n]
            else
                scale = 1.0f / (float)N_NODES;
#pragma unroll
            for (int q = 0; q < 2; ++q) {
                const int k = n * I_CH + q * 4 + half * 2;      // K quad base for this lane
                // A fragment: rows m=l16, K elems k, k+1 (8B aligned)
                v2f a = *(const v2f*)(Ub + (size_t)l16 * K_TOT + k);
                a.x *= scale; a.y *= scale;
                // B fragment: rows K=k (reg0), K=k+1 (reg1), column o=l16
                const float* wp = Wc + (size_t)k * O_CH + l16;
                v2f b; b.x = wp[0]; b.y = wp[O_CH];
                acc = __builtin_amdgcn_wmma_f32_16x16x4_f32(
                    false, a, false, b, (short)0, acc, false, false);
            }
        }
#pragma unroll
        for (int r = 0; r < 8; ++r) {                 // D: reg r -> row m=r(+8), col l16
            const int m = r + half * 8;
            red[wv][m * O_CH + l16] = acc[r];
        }
    };

    // ---- reduce partials across waves, squash, publish v to v_lds ----
    auto squash_to_v = [&]() {
        __syncthreads();
        float s = 0.0f;
#pragma unroll
        for (int w2 = 0; w2 < NWAVES; ++w2) s += red[w2][t];   // t = b*16 + o
        const float sn = half16_sum(s * s);                     // |s|^2 over o
        const float f  = sn / ((1.0f + sn) * sqrtf(sn));
        v_lds[t] = s * f;
        __syncthreads();
    };

    // ---- agreement pass: a[b,n] (+)= sum_o u_hat[b,n,o] * v[b,o] ----
    auto a_pass = [&](bool accumulate) {
        for (int n = n0; n < n0 + N_PER_WAVE; ++n) {
            v8f acc = {};
#pragma unroll
            for (int q = 0; q < 2; ++q) {
                const int k = n * I_CH + q * 4 + half * 2;
                v2f a = *(const v2f*)(Ub + (size_t)l16 * K_TOT + k);
                const float* wp = Wc + (size_t)k * O_CH + l16;
                v2f b; b.x = wp[0]; b.y = wp[O_CH];
                acc = __builtin_amdgcn_wmma_f32_16x16x4_f32(
                    false, a, false, b, (short)0, acc, false, false);
            }
#pragma unroll
            for (int r = 0; r < 8; ++r) {
                const int m = r + half * 8;
                float val = acc[r] * v_lds[m * O_CH + l16];     // u_hat * v, this o
                val = half16_sum(val);                          // sum over o
                if (l16 == 0) {
                    if (accumulate) a_lds[m * SN + n] += val;
                    else            a_lds[m * SN + n]  = val;
                }
            }
        }
        __syncthreads();
    };

    // ---- per-row softmax stats over n: amax_s[b], invz_s[b] ----
    auto softmax_stats = [&]() {
        const int b = t >> 4, j = t & 15;
        const float* row = a_lds + b * SN;
        float mx = -3.0e38f;
        for (int n = j; n < N_NODES; n += 16) mx = fmaxf(mx, row[n]);
        mx = half16_max(mx);
        float z = 0.0f;
        for (int n = j; n < N_NODES; n += 16) z += __expf(row[n] - mx);
        z = half16_sum(z);
        if (j == 0) { amax_s[b] = mx; invz_s[b] = 1.0f / z; }
        __syncthreads();
    };

    // ---------------- dynamic routing, 3 iterations ----------------
    gemm_pass(false);   // iter0: s0 = (1/N) * colsum_n(u_hat)
    squash_to_v();      // v0
    a_pass(false);      // b = a1
    softmax_stats();
    gemm_pass(true);    // iter1: s1 = sum_n softmax(b) * u_hat
    squash_to_v();      // v1
    a_pass(true);       // b = a1 + a2
    softmax_stats();
    gemm_pass(true);    // iter2: s2
    __syncthreads();

    // final squash + store v2 -> out[c, bb+b, 0, 0, o]
    {
        float s = 0.0f;
#pragma unroll
        for (int w2 = 0; w2 < NWAVES; ++w2) s += red[w2][t];
        const float sn = half16_sum(s * s);
        const float f  = sn / ((1.0f + sn) * sqrtf(sn));
        const int b = t >> 4, o = t & 15;
        out[(size_t)c * (B_SZ * O_CH) + (size_t)(bb + b) * O_CH + o] = s * f;
    }
}

extern "C" void kernel_launch(void* const* d_in, const int* in_sizes, int n_in,
                              void* d_out, int out_size, void* d_ws, size_t ws_size,
                              hipStream_t stream) {
    (void)in_sizes; (void)n_in; (void)d_ws; (void)ws_size; (void)out_size;
    const float* u = (const float*)d_in[0];   // [B, N, I] fp32
    const float* W = (const float*)d_in[1];   // [C, N, I, O] fp32
    float* out = (float*)d_out;               // [C, B, 1, 1, O] fp32

    dim3 grid(B_SZ / BT, C_CAPS);             // 32 x 10 = 320 workgroups
    digitcaps_routing_kernel<<<grid, NTHREADS, 0, stream>>>(u, W, out);
}